// TGCN_43903155699829
// MI455X (gfx1250) — compile-verified
//
#include <hip/hip_runtime.h>
#include <hip/hip_bf16.h>
#include <math.h>

// ---------------------------------------------------------------------------
// TGCN cell for MI455X (gfx1250, wave32).
// Bandwidth-bound (~150 MB traffic vs ~2.8 GFLOP): keep f32 end to end.
// GEMMs run on the matrix pipe via v_wmma_f32_16x16x4_f32 with both operands
// staged in LDS (padded-stride, bank-conflict-free ds_load_b64 in the loop).
// ---------------------------------------------------------------------------

typedef __attribute__((ext_vector_type(2))) float v2f;
typedef __attribute__((ext_vector_type(8))) float v8f;

#define TGCN_K 128  // fan_in = IN + OUT

// ---------------------------------------------------------------------------
// degree / normalization
// ---------------------------------------------------------------------------
__global__ void tgcn_deg_init(float* __restrict__ deg, int n) {
    int i = blockIdx.x * blockDim.x + threadIdx.x;
    if (i < n) deg[i] = 1.0f;  // self-loop weight
}

__global__ void tgcn_deg_acc(const long long* __restrict__ dst,
                             const float* __restrict__ w,
                             float* __restrict__ deg, int e) {
    int i = blockIdx.x * blockDim.x + threadIdx.x;
    if (i < e) atomicAdd(&deg[(int)dst[i]], w[i]);
}

__global__ void tgcn_deg_inv(float* __restrict__ deg, int n) {
    int i = blockIdx.x * blockDim.x + threadIdx.x;
    if (i < n) {
        float d = deg[i];
        deg[i] = (d > 0.0f) ? rsqrtf(fmaxf(d, 1e-12f)) : 0.0f;
    }
}

__global__ void tgcn_zero(float* __restrict__ p, int count) {
    int i = blockIdx.x * blockDim.x + threadIdx.x;
    if (i < count) p[i] = 0.0f;
}

// ---------------------------------------------------------------------------
// GEMM: Z[n x OUTC] = concat(A0, A1)[n x 128] @ W[128 x OUTC]
// Block = 4 waves = 16 rows x 64 cols; grid.y covers OUTC/64 column panels.
// Both operands staged in LDS with padded stride 132 floats:
//   - A tile loaded ONCE per block (not per wave)
//   - W tile stored transposed Wt[c][k] -> inner loop is pure
//     ds_load_b64 (immediate offsets) + v_wmma_f32_16x16x4_f32
// Lane layouts per CDNA5 ISA 7.12.2 (wave32):
//   A (16x4): lanes 0-15 -> M=lane, {K,K+1}; lanes 16-31 -> {K+2,K+3}
//   B (4x16): lanes 0-15 -> N=lane, rows {K,K+1}; lanes 16-31 rows {K+2,K+3}
//   C/D     : VGPR r -> M = r + 8*(lane>=16), N = lane%16
// ---------------------------------------------------------------------------
template <int OUTC>
__global__ void tgcn_gemm_wmma(const float* __restrict__ A0,  // n x 64
                               const float* __restrict__ A1,  // n x 64
                               const float* __restrict__ W,   // 128 x OUTC row-major
                               float* __restrict__ Z,         // n x OUTC
                               int n) {
    constexpr int PAD = 132;              // padded K stride (floats)
    __shared__ float At[16 * PAD];        // A tile  [r][k]
    __shared__ float Wt[64 * PAD];        // W tile  [c][k] (transposed)

    const int t    = threadIdx.x;         // 0..127
    const int lane = t & 31;
    const int wave = t >> 5;              // 0..3
    const int l    = lane & 15;
    const int half = lane >> 4;

    const int row0 = blockIdx.x * 16;
    const int cb0  = blockIdx.y * 64;     // first output column of this block

    // ---- stage A tile (16 rows x 128 k) : 2048 floats, 16 per thread ----
    for (int i = 0; i < 16; ++i) {
        int lin = t + i * 128;
        int r = lin >> 7, k = lin & 127;
        int row = row0 + r;
        if (row >= n) row = n - 1;        // clamp loads; stores guarded below
        At[r * PAD + k] =
            (k < 64) ? A0[row * 64 + k] : A1[row * 64 + (k - 64)];
    }
    // ---- stage W tile transposed (64 cols x 128 k) : 8192 floats ----
    for (int i = 0; i < 64; ++i) {
        int lin = t + i * 128;
        int c = lin & 63, k = lin >> 6;   // coalesced global read across c
        Wt[c * PAD + k] = W[k * OUTC + cb0 + c];
    }
    __syncthreads();

    // 8-byte aligned fragment base pointers (PAD even, 2*half even)
    const v2f* ap = (const v2f*)(At + (l * PAD + 2 * half));
    const v2f* bp = (const v2f*)(Wt + ((wave * 16 + l) * PAD + 2 * half));

    v8f acc = {};
    #pragma unroll
    for (int kk = 0; kk < TGCN_K / 4; ++kk) {
        v2f a = ap[2 * kk];               // ds_load_b64, immediate offset
        v2f b = bp[2 * kk];               // ds_load_b64, immediate offset
        acc = __builtin_amdgcn_wmma_f32_16x16x4_f32(
            /*neg_a=*/false, a, /*neg_b=*/false, b,
            /*c_mod=*/(short)0, acc, /*reuse_a=*/false, /*reuse_b=*/false);
    }

    const int col = cb0 + wave * 16 + l;
    #pragma unroll
    for (int r = 0; r < 8; ++r) {
        const int row = row0 + r + half * 8;
        if (row < n) Z[row * OUTC + col] = acc[r];
    }
}

// ---------------------------------------------------------------------------
// Edge aggregation: out[dst] += dinv[src]*w*dinv[dst] * Z[src]
// One wave per edge; float4 source traffic; atomic scatter into dst row.
// ---------------------------------------------------------------------------
__global__ void tgcn_agg_edges(const long long* __restrict__ src,
                               const long long* __restrict__ dst,
                               const float* __restrict__ w,
                               const float* __restrict__ dinv,
                               const float* __restrict__ Z,
                               float* __restrict__ out,
                               int e, int cols) {
    const int warp = (int)(((size_t)blockIdx.x * blockDim.x + threadIdx.x) >> 5);
    const int lane = threadIdx.x & 31;
    if (warp >= e) return;

    const int s = (int)src[warp];
    const int d = (int)dst[warp];
    const float norm = dinv[s] * w[warp] * dinv[d];

    const float4* zs = (const float4*)(Z + s * cols);
    float* od = out + d * cols;
    __builtin_prefetch(zs, 0, 1);  // global_prefetch_b8 of the source row

    const int q = cols >> 2;
    for (int j = lane; j < q; j += 32) {
        float4 v = zs[j];
        atomicAdd(od + 4 * j + 0, norm * v.x);
        atomicAdd(od + 4 * j + 1, norm * v.y);
        atomicAdd(od + 4 * j + 2, norm * v.z);
        atomicAdd(od + 4 * j + 3, norm * v.w);
    }
}

// ---------------------------------------------------------------------------
// Pointwise stage 1: ru = sigmoid(agg1 + dinv^2 * Z1 + b1)
//   j <  64: r -> Hr = H * r ;  j >= 64: u -> U
// (self-loop term dinv^2 * Z1 folded in here; avoids a second scatter pass)
// ---------------------------------------------------------------------------
__global__ void tgcn_pw1(const float* __restrict__ agg1,
                         const float* __restrict__ Z1,
                         const float* __restrict__ dinv,
                         const float* __restrict__ H,
                         const float* __restrict__ b1,
                         float* __restrict__ U,
                         float* __restrict__ Hr,
                         int n) {
    const int i = blockIdx.x * blockDim.x + threadIdx.x;
    if (i >= n * 128) return;
    const int node = i >> 7;
    const int j    = i & 127;
    const float di = dinv[node];
    const float x  = agg1[i] + di * di * Z1[i] + b1[j];
    const float sg = 1.0f / (1.0f + expf(-x));
    const int k = node * 64 + (j & 63);
    if (j < 64) Hr[k] = H[k] * sg;
    else        U[k]  = sg;
}

// ---------------------------------------------------------------------------
// Pointwise stage 2: c = tanh(agg2 + dinv^2 * Z2 + b2); out = u*H + (1-u)*c
// ---------------------------------------------------------------------------
__global__ void tgcn_pw2(const float* __restrict__ agg2,
                         const float* __restrict__ Z2,
                         const float* __restrict__ dinv,
                         const float* __restrict__ H,
                         const float* __restrict__ b2,
                         const float* __restrict__ U,
                         float* __restrict__ out,
                         int n) {
    const int i = blockIdx.x * blockDim.x + threadIdx.x;
    if (i >= n * 64) return;
    const int node = i >> 6;
    const int j    = i & 63;
    const float di = dinv[node];
    const float c  = tanhf(agg2[i] + di * di * Z2[i] + b2[j]);
    const float u  = U[i];
    out[i] = u * H[i] + (1.0f - u) * c;
}

// ---------------------------------------------------------------------------
// launch
// ---------------------------------------------------------------------------
extern "C" void kernel_launch(void* const* d_in, const int* in_sizes, int n_in,
                              void* d_out, int out_size, void* d_ws, size_t ws_size,
                              hipStream_t stream) {
    const float*     X  = (const float*)d_in[0];      // n x 64
    const float*     H  = (const float*)d_in[1];      // n x 64
    const long long* EI = (const long long*)d_in[2];  // 2 x e (int64)
    const float*     EW = (const float*)d_in[3];      // e
    const float*     W1 = (const float*)d_in[4];      // 128 x 128
    const float*     B1 = (const float*)d_in[5];      // 128
    const float*     W2 = (const float*)d_in[6];      // 128 x 64
    const float*     B2 = (const float*)d_in[7];      // 64

    const int n = in_sizes[0] / 64;
    const int e = in_sizes[3];
    const long long* src = EI;
    const long long* dst = EI + e;

    // workspace carve (floats): dinv | Z1 | agg1 | U | Hr ; Z2/agg2 alias Z1/agg1
    float* ws   = (float*)d_ws;
    float* dinv = ws;                      // n
    float* Z1   = dinv + n;                // n*128
    float* agg1 = Z1 + (size_t)n * 128;    // n*128
    float* U    = agg1 + (size_t)n * 128;  // n*64
    float* Hr   = U + (size_t)n * 64;      // n*64
    float* Z2   = Z1;                      // Z1 dead after pw1
    float* agg2 = agg1;                    // agg1 dead after pw1
    float* outp = (float*)d_out;

    const int T = 256;
    const int rowTiles = (n + 15) / 16;

    // 1) normalization coefficients
    tgcn_deg_init<<<(n + T - 1) / T, T, 0, stream>>>(dinv, n);
    tgcn_deg_acc <<<(e + T - 1) / T, T, 0, stream>>>(dst, EW, dinv, e);
    tgcn_deg_inv <<<(n + T - 1) / T, T, 0, stream>>>(dinv, n);

    // 2) GCN layer 1: Z1 = [X,H] @ W1 ; agg1 = scatter(Z1) ; ru = sigmoid(. + b1)
    tgcn_zero<<<(n * 128 + T - 1) / T, T, 0, stream>>>(agg1, n * 128);
    tgcn_gemm_wmma<128><<<dim3(rowTiles, 2), 128, 0, stream>>>(X, H, W1, Z1, n);
    tgcn_agg_edges<<<(e + 7) / 8, T, 0, stream>>>(src, dst, EW, dinv, Z1, agg1, e, 128);
    tgcn_pw1<<<(n * 128 + T - 1) / T, T, 0, stream>>>(agg1, Z1, dinv, H, B1, U, Hr, n);

    // 3) GCN layer 2: Z2 = [X, H*r] @ W2 ; agg2 = scatter(Z2) ; c = tanh(. + b2)
    tgcn_zero<<<(n * 64 + T - 1) / T, T, 0, stream>>>(agg2, n * 64);
    tgcn_gemm_wmma<64><<<dim3(rowTiles, 1), 128, 0, stream>>>(X, Hr, W2, Z2, n);
    tgcn_agg_edges<<<(e + 7) / 8, T, 0, stream>>>(src, dst, EW, dinv, Z2, agg2, e, 64);

    // 4) gate
    tgcn_pw2<<<(n * 64 + T - 1) / T, T, 0, stream>>>(agg2, Z2, dinv, H, B2, U, outp, n);
}